// MultiHeadedSelfAttention_51402168598693
// MI455X (gfx1250) — compile-verified
//
#include <hip/hip_runtime.h>

// ---------------- CDNA5 (gfx1250) f32 WMMA attention ----------------
// D = A(16x4 f32) x B(4x16 f32) + C(16x16 f32), wave32.
typedef float v2f __attribute__((ext_vector_type(2)));
typedef float v8f __attribute__((ext_vector_type(8)));

#define TLEN 4096
#define DIM_ 1024
#define NH_  16
#define HD_  64
#define QKVLD 3072   // row stride of qkv buffer (3*DIM)

__device__ __forceinline__ v8f v8f_zero() {
    v8f z = {0.f, 0.f, 0.f, 0.f, 0.f, 0.f, 0.f, 0.f};
    return z;
}

__device__ __forceinline__ v8f wmma_f32(v2f a, v2f b, v8f c) {
    // 8 args: (neg_a, A, neg_b, B, c_mod, C, reuse_a, reuse_b)
    return __builtin_amdgcn_wmma_f32_16x16x4_f32(false, a, false, b, (short)0, c,
                                                 false, false);
}

// --- CDNA5 async global->LDS copy (ASYNCcnt-tracked, bypasses VGPRs) ---
// LDS byte address = low 32 bits of the generic pointer (ISA 10.2 aperture map).
__device__ __forceinline__ void async_copy16(void* lds, const void* gptr) {
    unsigned l = (unsigned)(unsigned long long)(uintptr_t)lds;
    unsigned long long g = (unsigned long long)(uintptr_t)gptr;
    asm volatile("global_load_async_to_lds_b128 %0, %1, off" ::"v"(l), "v"(g)
                 : "memory");
}
template <int N>
__device__ __forceinline__ void wait_async() {
    asm volatile("s_wait_asynccnt %0" ::"n"(N) : "memory");
}

// A-fragment (and B-fragment from [N][K]-layout LDS, identical access):
// lane<16 holds K = {4kk, 4kk+1}; lane>=16 holds K = {4kk+2, 4kk+3}; row/col = row0 + (lane&15)
__device__ __forceinline__ v2f frag_nk(const float* lds, int row0, int kk, int ld) {
    int lane = threadIdx.x & 31;
    return *(const v2f*)(lds + (row0 + (lane & 15)) * ld + kk * 4 + 2 * (lane >> 4));
}

// B-fragment from [K][N]-layout LDS (V tiles): b = { B[kb][n], B[kb+1][n] }
__device__ __forceinline__ v2f frag_kn(const float* lds, int n0, int kk, int ld) {
    int lane = threadIdx.x & 31;
    int kb = kk * 4 + 2 * (lane >> 4);
    int n = n0 + (lane & 15);
    v2f b;
    b.x = lds[kb * ld + n];
    b.y = lds[(kb + 1) * ld + n];
    return b;
}

// Monotone float<->uint key so unsigned atomicMax == float max (exact, order-independent).
__device__ __forceinline__ unsigned fkey(float f) {
    unsigned u = __float_as_uint(f);
    return (u & 0x80000000u) ? ~u : (u | 0x80000000u);
}
__device__ __forceinline__ float keyf(unsigned k) {
    unsigned u = (k & 0x80000000u) ? (k ^ 0x80000000u) : ~k;
    return __uint_as_float(u);
}

__global__ void init_kernel(unsigned* maxkey) {
    if (threadIdx.x < NH_) maxkey[threadIdx.x] = 0u;  // key(0) <= every real key
}

// ------------------------------------------------------------------
// C[M,N] = A[M,K] @ B[N,K]^T   (torch Linear layout)
// Block tile 128x128, BK=16, 256 threads = 8 waves (4 M x 2 N),
// wave tile 32x64 = 2x4 accumulators. Double-buffered async global->LDS:
// next k-tile's copies are in flight while the current tile is computed.
// ------------------------------------------------------------------
__global__ __launch_bounds__(256) void gemm_nt_wmma(const float* __restrict__ A,
                                                    const float* __restrict__ B,
                                                    float* __restrict__ C, int M, int N,
                                                    int K) {
    const int LD = 20;  // 16 + 4 pad: 16B-aligned float4 rows, conflict-free b64 frags
    __shared__ __align__(16) float lA[2][128 * 20];
    __shared__ __align__(16) float lB[2][128 * 20];

    int tid = threadIdx.x;
    int wave = tid >> 5, lane = tid & 31;
    int lm = lane & 15, lh = lane >> 4;
    int mBase = blockIdx.y * 128, nBase = blockIdx.x * 128;
    int wm = wave & 3, wn = wave >> 2;

    // this thread's 4 transfers per k-tile (2 for A, 2 for B)
    auto issue = [&](int buf, int k0) {
        for (int i = 0; i < 2; i++) {
            int idx = tid + i * 256;  // 0..511 over 128x16 tile, float4 granules
            int r = idx >> 2, c4 = idx & 3;
            async_copy16(&lA[buf][r * LD + c4 * 4],
                         A + (size_t)(mBase + r) * K + k0 + c4 * 4);
        }
        for (int i = 0; i < 2; i++) {
            int idx = tid + i * 256;
            int r = idx >> 2, c4 = idx & 3;
            async_copy16(&lB[buf][r * LD + c4 * 4],
                         B + (size_t)(nBase + r) * K + k0 + c4 * 4);
        }
    };

    v8f acc[2][4];
    for (int i = 0; i < 2; i++)
        for (int j = 0; j < 4; j++) acc[i][j] = v8f_zero();

    int ntiles = K >> 4;
    issue(0, 0);
    for (int t = 0; t < ntiles; t++) {
        int cur = t & 1;
        if (t + 1 < ntiles) {
            issue(cur ^ 1, (t + 1) << 4);  // overlap next tile with this compute
            wait_async<4>();  // async loads retire in order: oldest 4 (tile t) done
        } else {
            wait_async<0>();
        }
        __syncthreads();
        const float* la = lA[cur];
        const float* lb = lB[cur];
        for (int kk = 0; kk < 4; kk++) {
            v2f bf[4], af[2];
            for (int j = 0; j < 4; j++) bf[j] = frag_nk(lb, wn * 64 + j * 16, kk, LD);
            for (int i = 0; i < 2; i++) af[i] = frag_nk(la, wm * 32 + i * 16, kk, LD);
            for (int i = 0; i < 2; i++)
                for (int j = 0; j < 4; j++) acc[i][j] = wmma_f32(af[i], bf[j], acc[i][j]);
        }
        __syncthreads();  // everyone done with buf `cur` before t+1 overwrites it
    }

    for (int i = 0; i < 2; i++)
        for (int j = 0; j < 4; j++) {
            int r0 = mBase + wm * 32 + i * 16 + 8 * lh;
            int c = nBase + wn * 64 + j * 16 + lm;
            for (int v = 0; v < 8; v++) C[(size_t)(r0 + v) * N + c] = acc[i][j][v];
        }
}

// ------------------------------------------------------------------
// Pass 1: per-head global max of scale * (Q K^T). Block = (head, 64 q-rows),
// loops s in blocks of 128. 8 waves: wave = rows 16*(w&3), cols 64*(w>>2).
// ------------------------------------------------------------------
__global__ __launch_bounds__(256) void attn_max_kernel(const float* __restrict__ qkv,
                                                       unsigned* __restrict__ maxkey) {
    const int LDQ = 68, LDK = 68;
    __shared__ __align__(16) float lQ[64 * 68];
    __shared__ __align__(16) float lK[128 * 68];
    __shared__ float red[256];

    int h = blockIdx.y, qb = blockIdx.x;
    int tid = threadIdx.x, wave = tid >> 5;
    int wm = wave & 3, wn = wave >> 2;

    const float* Q = qkv + (size_t)h * HD_;
    const float* Kp = qkv + DIM_ + (size_t)h * HD_;
    int q0 = qb * 64;

    for (int i = 0; i < 4; i++) {  // Q block: 64x64 = 1024 float4
        int idx = tid + i * 256;
        int r = idx >> 4, c4 = idx & 15;
        async_copy16(&lQ[r * LDQ + c4 * 4],
                     Q + (size_t)(q0 + r) * QKVLD + c4 * 4);
    }

    float m = -3.0e38f;
    for (int s0 = 0; s0 < TLEN; s0 += 128) {
        __syncthreads();  // prior compute done with lK
        for (int i = 0; i < 8; i++) {  // K block: 128x64 = 2048 float4
            int idx = tid + i * 256;
            int r = idx >> 4, c4 = idx & 15;
            async_copy16(&lK[r * LDK + c4 * 4],
                         Kp + (size_t)(s0 + r) * QKVLD + c4 * 4);
        }
        wait_async<0>();
        __syncthreads();
        v8f acc[4];
        for (int j = 0; j < 4; j++) acc[j] = v8f_zero();
        for (int kk = 0; kk < 16; kk++) {
            v2f a = frag_nk(lQ, 16 * wm, kk, LDQ);
            for (int j = 0; j < 4; j++) {
                v2f b = frag_nk(lK, 64 * wn + 16 * j, kk, LDK);
                acc[j] = wmma_f32(a, b, acc[j]);
            }
        }
        for (int j = 0; j < 4; j++)
            for (int v = 0; v < 8; v++) m = fmaxf(m, acc[j][v]);
    }
    m *= 0.125f;  // HEAD_DIM^-0.5, positive => order preserved

    red[tid] = m;
    __syncthreads();
    for (int s = 128; s > 0; s >>= 1) {
        if (tid < s) red[tid] = fmaxf(red[tid], red[tid + s]);
        __syncthreads();
    }
    if (tid == 0) atomicMax(&maxkey[h], fkey(red[0]));
}

// ------------------------------------------------------------------
// Pass 2: O_unnorm = exp(scale*QK^T - M[h]) @ V, plus per-block sum of exp.
// Block = (head, 64 q-rows), s-blocks of 64. K/V share one LDS buffer
// (K in phase 1, V in phase 2). P staged in LDS between the two WMMA GEMMs.
// ------------------------------------------------------------------
__global__ __launch_bounds__(256) void attn_pv_kernel(const float* __restrict__ qkv,
                                                      const unsigned* __restrict__ maxkey,
                                                      float* __restrict__ attnU,
                                                      float* __restrict__ partial) {
    const int LDQ = 68, LDKV = 68, LDP = 66;
    __shared__ __align__(16) float lQ[64 * 68];
    __shared__ __align__(16) float lKV[64 * 68];
    __shared__ __align__(16) float lP[64 * 66];
    __shared__ float red[256];

    int h = blockIdx.y, qb = blockIdx.x;
    int tid = threadIdx.x, wave = tid >> 5, lane = tid & 31;
    int lm = lane & 15, lh = lane >> 4;
    int wm = wave & 3, wn = wave >> 2;  // rows 16*wm, cols 32*wn (2 tiles)

    const float M = keyf(maxkey[h]);
    const float* Q = qkv + (size_t)h * HD_;
    const float* Kp = qkv + DIM_ + (size_t)h * HD_;
    const float* Vp = qkv + 2 * DIM_ + (size_t)h * HD_;
    int q0 = qb * 64;

    for (int i = 0; i < 4; i++) {
        int idx = tid + i * 256;
        int r = idx >> 4, c4 = idx & 15;
        async_copy16(&lQ[r * LDQ + c4 * 4],
                     Q + (size_t)(q0 + r) * QKVLD + c4 * 4);
    }

    v8f o[2];
    o[0] = v8f_zero();
    o[1] = v8f_zero();
    float psum = 0.f;

    for (int s0 = 0; s0 < TLEN; s0 += 64) {
        __syncthreads();  // prior iter done with lKV(V) and lP
        for (int i = 0; i < 4; i++) {
            int idx = tid + i * 256;
            int r = idx >> 4, c4 = idx & 15;
            async_copy16(&lKV[r * LDKV + c4 * 4],
                         Kp + (size_t)(s0 + r) * QKVLD + c4 * 4);
        }
        wait_async<0>();
        __syncthreads();
        // S = Q @ K^T
        v8f s[2];
        s[0] = v8f_zero();
        s[1] = v8f_zero();
        for (int kk = 0; kk < 16; kk++) {
            v2f a = frag_nk(lQ, 16 * wm, kk, LDQ);
            for (int j = 0; j < 2; j++) {
                v2f b = frag_nk(lKV, 32 * wn + 16 * j, kk, LDKV);
                s[j] = wmma_f32(a, b, s[j]);
            }
        }
        __syncthreads();  // all waves done reading K before V overwrites lKV
        // V tile into shared K/V buffer (async, overlaps the exp work below)
        for (int i = 0; i < 4; i++) {
            int idx = tid + i * 256;
            int r = idx >> 4, c4 = idx & 15;
            async_copy16(&lKV[r * LDKV + c4 * 4],
                         Vp + (size_t)(s0 + r) * QKVLD + c4 * 4);
        }
        // P = exp(scale*S - M) -> LDS; accumulate exp-sum
        for (int j = 0; j < 2; j++) {
            int r0 = 16 * wm + 8 * lh;
            int c0 = 32 * wn + 16 * j + lm;
            for (int v = 0; v < 8; v++) {
                float p = __expf(s[j][v] * 0.125f - M);
                psum += p;
                lP[(r0 + v) * LDP + c0] = p;
            }
        }
        wait_async<0>();
        __syncthreads();  // P visible, V loaded
        // O += P @ V
        for (int kk = 0; kk < 16; kk++) {
            v2f a = frag_nk(lP, 16 * wm, kk, LDP);
            for (int j = 0; j < 2; j++) {
                v2f b = frag_kn(lKV, 32 * wn + 16 * j, kk, LDKV);
                o[j] = wmma_f32(a, b, o[j]);
            }
        }
    }

    // Unnormalized O -> attnU[T][DIM] at this head's column block
    for (int j = 0; j < 2; j++) {
        int r0 = q0 + 16 * wm + 8 * lh;
        int c = h * HD_ + 32 * wn + 16 * j + lm;
        for (int v = 0; v < 8; v++) attnU[(size_t)(r0 + v) * DIM_ + c] = o[j][v];
    }

    // Deterministic partial exp-sum per (head, q-block)
    red[tid] = psum;
    __syncthreads();
    for (int s = 128; s > 0; s >>= 1) {
        if (tid < s) red[tid] += red[tid + s];
        __syncthreads();
    }
    if (tid == 0) partial[h * 64 + qb] = red[0];
}

__global__ void denom_reduce_kernel(const float* __restrict__ partial,
                                    float* __restrict__ denom) {
    __shared__ float r[64];
    r[threadIdx.x] = partial[blockIdx.x * 64 + threadIdx.x];
    __syncthreads();
    for (int s = 32; s > 0; s >>= 1) {
        if (threadIdx.x < s) r[threadIdx.x] += r[threadIdx.x + s];
        __syncthreads();
    }
    if (threadIdx.x == 0) denom[blockIdx.x] = r[0];
}

// Normalize attnU in place: column block c belongs to head c>>6.
__global__ __launch_bounds__(256) void scale_attn_kernel(float* __restrict__ attnU,
                                                         const float* __restrict__ denom) {
    __shared__ float rs[16];
    if (threadIdx.x < 16) rs[threadIdx.x] = 1.0f / denom[threadIdx.x];
    __syncthreads();
    size_t i = (size_t)blockIdx.x * 256 + threadIdx.x;  // float4 index, 256 per row
    float4* p = (float4*)attnU;
    float4 v = p[i];
    float s = rs[(int)(i & 255) >> 4];  // (c4*4)>>6
    v.x *= s;
    v.y *= s;
    v.z *= s;
    v.w *= s;
    p[i] = v;
}

// ------------------------------------------------------------------
extern "C" void kernel_launch(void* const* d_in, const int* in_sizes, int n_in,
                              void* d_out, int out_size, void* d_ws, size_t ws_size,
                              hipStream_t stream) {
    const float* x = (const float*)d_in[0];       // [4096,1024]
    const float* W_in = (const float*)d_in[1];    // [3072,1024]
    const float* W_out = (const float*)d_in[2];   // [1024,1024]
    float* out = (float*)d_out;                   // [4096,1024]

    // workspace: qkv (50.3MB) | attnU (16.8MB) | denom[16] | partial[16*64] | maxkey[16]
    float* qkv = (float*)d_ws;
    float* attnU = qkv + (size_t)TLEN * 3 * DIM_;
    float* denom = attnU + (size_t)TLEN * DIM_;
    float* partial = denom + 16;
    unsigned* maxkey = (unsigned*)(partial + 16 * 64);

    init_kernel<<<1, 32, 0, stream>>>(maxkey);

    dim3 g1(3072 / 128, 4096 / 128);
    gemm_nt_wmma<<<g1, 256, 0, stream>>>(x, W_in, qkv, 4096, 3072, 1024);

    dim3 ga(64, 16);  // (q-block, head)
    attn_max_kernel<<<ga, 256, 0, stream>>>(qkv, maxkey);
    attn_pv_kernel<<<ga, 256, 0, stream>>>(qkv, maxkey, attnU, partial);
    denom_reduce_kernel<<<16, 64, 0, stream>>>(partial, denom);

    // normalize, then output projection
    scale_attn_kernel<<<(TLEN * DIM_ / 4) / 256, 256, 0, stream>>>(attnU, denom);
    dim3 g4(1024 / 128, 4096 / 128);
    gemm_nt_wmma<<<g4, 256, 0, stream>>>(attnU, W_out, out, 4096, 1024, 1024);
}